// PatchDecoder_33603824124664
// MI455X (gfx1250) — compile-verified
//
#include <hip/hip_runtime.h>
#include <math.h>

typedef __attribute__((ext_vector_type(2))) float v2f;
typedef __attribute__((ext_vector_type(8))) float v8f;

#define HDIM 2048
#define SEQ  32
#define NHEAD 8
#define DHEAD 256
#define NLAYER 3

__device__ __forceinline__ float gelu_exact(float x) {
  // exact gelu: 0.5*x*(1+erf(x/sqrt(2)))
  return 0.5f * x * (1.0f + erff(x * 0.70710678118654752440f));
}

// ---------------------------------------------------------------------------
// CDNA5 async copy: global -> LDS, 16 bytes per lane, tracked by ASYNCcnt.
// LDS byte address = low 32 bits of the generic shared pointer (ISA 10.2).
// ---------------------------------------------------------------------------
__device__ __forceinline__ void async_copy_b128(const float* gptr,
                                                const float* lds_ptr) {
  const unsigned lds = (unsigned)(unsigned long long)lds_ptr;
  asm volatile("global_load_async_to_lds_b128 %0, %1, off"
               :
               : "v"(lds), "v"(gptr)
               : "memory");
}

__device__ __forceinline__ void wait_async0() {
  asm volatile("s_wait_asynccnt 0x0" ::: "memory");
}

// ---------------------------------------------------------------------------
// GEMM: out[32,N] = act(X[32,K] @ W[N,K]^T + bias[N])
// grid.x = N/128, block = 256 threads (8 wave32s, 16 output columns per wave)
// V_WMMA_F32_16X16X4_F32 + double-buffered LDS fed by async-to-LDS copies.
// ---------------------------------------------------------------------------
template <int ACT>  // 0 = none, 1 = exact gelu
__global__ __launch_bounds__(256) void gemm32_wmma(const float* __restrict__ X,
                                                   const float* __restrict__ W,
                                                   const float* __restrict__ bias,
                                                   float* __restrict__ out,
                                                   int K, int N) {
  constexpr int BK  = 32;
  constexpr int LDW = BK + 4;  // 36 floats: keeps 16B alignment, spreads banks
  __shared__ float Xs[2][32 * LDW];
  __shared__ float Ws[2][128 * LDW];

  const int tid  = threadIdx.x;
  const int wave = tid >> 5;
  const int lane = tid & 31;
  const int lo   = lane & 15;   // within-halfwave index
  const int hi   = lane >> 4;   // halfwave select
  const int n0   = blockIdx.x * 128;

  const int seg  = (tid & 7) * 4;  // float offset of this thread's 16B slice
  const int rowx = tid >> 3;       // 0..31

  v8f c_lo = {};  // rows 0..15 of the 32-row tile
  v8f c_hi = {};  // rows 16..31

  // ---- prologue: async-stage chunk 0 into buffer 0 ----
  {
    async_copy_b128(X + (size_t)rowx * K + seg, &Xs[0][rowx * LDW + seg]);
#pragma unroll
    for (int p = 0; p < 4; ++p) {
      const int r = rowx + 32 * p;
      async_copy_b128(W + (size_t)(n0 + r) * K + seg, &Ws[0][r * LDW + seg]);
    }
  }
  wait_async0();
  __syncthreads();

  int buf = 0;
  for (int k0 = 0; k0 < K; k0 += BK) {
    // ---- async-stage next chunk into the other buffer (overlaps WMMAs) ----
    if (k0 + BK < K) {
      const int nb = buf ^ 1;
      async_copy_b128(X + (size_t)rowx * K + k0 + BK + seg,
                      &Xs[nb][rowx * LDW + seg]);
#pragma unroll
      for (int p = 0; p < 4; ++p) {
        const int r = rowx + 32 * p;
        async_copy_b128(W + (size_t)(n0 + r) * K + k0 + BK + seg,
                        &Ws[nb][r * LDW + seg]);
      }
    }

    // ---- 8 x WMMA_F32_16X16X4_F32 per accumulator over this K chunk ----
    const float* xs = &Xs[buf][0];
    const float* ws = &Ws[buf][0];
#pragma unroll
    for (int kk = 0; kk < BK; kk += 4) {
      // A 16x4 f32 layout: lanes 0-15 hold K=kk,kk+1; lanes 16-31 K=kk+2,kk+3
      v2f a_lo, a_hi, b;
      a_lo.x = xs[lo * LDW + kk + 2 * hi];
      a_lo.y = xs[lo * LDW + kk + 2 * hi + 1];
      a_hi.x = xs[(16 + lo) * LDW + kk + 2 * hi];
      a_hi.y = xs[(16 + lo) * LDW + kk + 2 * hi + 1];
      // B 4x16 f32: VGPR0 rows K=kk (lanes0-15)/kk+1 (lanes16-31), VGPR1 +2
      const int c = (wave * 16 + lo) * LDW;
      b.x = ws[c + kk + hi];
      b.y = ws[c + kk + 2 + hi];
      c_lo = __builtin_amdgcn_wmma_f32_16x16x4_f32(false, a_lo, false, b,
                                                   (short)0, c_lo, false, false);
      c_hi = __builtin_amdgcn_wmma_f32_16x16x4_f32(false, a_hi, false, b,
                                                   (short)0, c_hi, false, false);
    }

    // ---- make next buffer visible to everyone, then swap ----
    wait_async0();
    __syncthreads();
    buf ^= 1;
  }

  // ---- epilogue: C/D layout — VGPR j: lanes0-15 -> M=j, lanes16-31 -> M=j+8
  const int n  = n0 + wave * 16 + lo;
  const float bv = bias ? bias[n] : 0.0f;
#pragma unroll
  for (int j = 0; j < 8; ++j) {
    const int m = j + 8 * hi;
    float r0 = c_lo[j] + bv;
    float r1 = c_hi[j] + bv;
    if (ACT == 1) { r0 = gelu_exact(r0); r1 = gelu_exact(r1); }
    out[(size_t)m * N + n]        = r0;
    out[(size_t)(m + 16) * N + n] = r1;
  }
}

// ---------------------------------------------------------------------------
// GEMV: out[n] = act(dot(x, W[n,:]) + bias[n]); one wave32 per output row.
// grid.x = ceil(N/8), block = 256 (8 waves)
// ---------------------------------------------------------------------------
__global__ __launch_bounds__(256) void gemv_wt(const float* __restrict__ x,
                                               const float* __restrict__ W,
                                               const float* __restrict__ bias,
                                               float* __restrict__ out,
                                               int K, int N, int act) {
  const int wave = threadIdx.x >> 5;
  const int lane = threadIdx.x & 31;
  const int n = blockIdx.x * 8 + wave;
  if (n >= N) return;
  const float* wr = W + (size_t)n * K;
  float acc = 0.0f;
  for (int base = lane * 4; base < K; base += 128) {
    const float4 wv = *reinterpret_cast<const float4*>(wr + base);
    const float4 xv = *reinterpret_cast<const float4*>(x + base);
    acc += wv.x * xv.x + wv.y * xv.y + wv.z * xv.z + wv.w * xv.w;
    __builtin_prefetch(wr + base + 128, 0, 3);
  }
#pragma unroll
  for (int off = 16; off > 0; off >>= 1) acc += __shfl_down(acc, off, 32);
  if (lane == 0) {
    float r = acc + (bias ? bias[n] : 0.0f);
    if (act) r = gelu_exact(r);
    out[n] = r;
  }
}

// ---------------------------------------------------------------------------
// Self-attention over qkv[32, 3H] (q|k|v packed): one block per head.
// ---------------------------------------------------------------------------
__global__ __launch_bounds__(256) void attn_self(const float* __restrict__ qkv,
                                                 float* __restrict__ att_o) {
  __shared__ float Ssc[32 * 33];
  const int h = blockIdx.x;
  const int tid = threadIdx.x;
  const int RS = 3 * HDIM;
  const float* q = qkv + h * DHEAD;
  const float* k = qkv + HDIM + h * DHEAD;
  const float* v = qkv + 2 * HDIM + h * DHEAD;

  for (int idx = tid; idx < 32 * 32; idx += 256) {
    const int i = idx >> 5, j = idx & 31;
    const float* qi = q + (size_t)i * RS;
    const float* kj = k + (size_t)j * RS;
    float s = 0.0f;
    for (int d = 0; d < DHEAD; d += 4) {
      const float4 a = *reinterpret_cast<const float4*>(qi + d);
      const float4 b = *reinterpret_cast<const float4*>(kj + d);
      s += a.x * b.x + a.y * b.y + a.z * b.z + a.w * b.w;
    }
    Ssc[i * 33 + j] = s * 0.0625f;  // 1/sqrt(256)
  }
  __syncthreads();
  if (tid < 32) {
    float mx = -1e30f;
    for (int j = 0; j < 32; ++j) mx = fmaxf(mx, Ssc[tid * 33 + j]);
    float sum = 0.0f;
    for (int j = 0; j < 32; ++j) {
      const float e = expf(Ssc[tid * 33 + j] - mx);
      Ssc[tid * 33 + j] = e;
      sum += e;
    }
    const float inv = 1.0f / sum;
    for (int j = 0; j < 32; ++j) Ssc[tid * 33 + j] *= inv;
  }
  __syncthreads();
  for (int idx = tid; idx < 32 * DHEAD; idx += 256) {
    const int i = idx >> 8, d = idx & 255;
    float s = 0.0f;
    for (int j = 0; j < 32; ++j) s += Ssc[i * 33 + j] * v[(size_t)j * RS + d];
    att_o[(size_t)i * HDIM + h * DHEAD + d] = s;
  }
}

// ---------------------------------------------------------------------------
// x[row] = LayerNorm(x[row] + t[row*tstride]) * g + b   (tstride 0 => broadcast)
// grid = 32 rows, block = 256
// ---------------------------------------------------------------------------
__global__ __launch_bounds__(256) void resid_ln(float* __restrict__ x,
                                                const float* __restrict__ t,
                                                int tstride,
                                                const float* __restrict__ g,
                                                const float* __restrict__ b) {
  __shared__ float red[8];
  __shared__ float s_mean, s_rstd;
  const int r = blockIdx.x, tid = threadIdx.x;
  const float* trow = t + (size_t)r * tstride;
  float y[8];
  float sum = 0.0f;
#pragma unroll
  for (int i = 0; i < 8; ++i) {
    const int d = tid + i * 256;
    y[i] = x[(size_t)r * HDIM + d] + trow[d];
    sum += y[i];
  }
#pragma unroll
  for (int off = 16; off > 0; off >>= 1) sum += __shfl_down(sum, off, 32);
  if ((tid & 31) == 0) red[tid >> 5] = sum;
  __syncthreads();
  if (tid == 0) {
    float s = 0.0f;
    for (int w = 0; w < 8; ++w) s += red[w];
    s_mean = s * (1.0f / HDIM);
  }
  __syncthreads();
  const float mean = s_mean;
  float vs = 0.0f;
#pragma unroll
  for (int i = 0; i < 8; ++i) {
    const float d = y[i] - mean;
    vs += d * d;
  }
#pragma unroll
  for (int off = 16; off > 0; off >>= 1) vs += __shfl_down(vs, off, 32);
  if ((tid & 31) == 0) red[tid >> 5] = vs;
  __syncthreads();
  if (tid == 0) {
    float s = 0.0f;
    for (int w = 0; w < 8; ++w) s += red[w];
    s_rstd = rsqrtf(s * (1.0f / HDIM) + 1e-5f);
  }
  __syncthreads();
  const float rstd = s_rstd;
#pragma unroll
  for (int i = 0; i < 8; ++i) {
    const int d = tid + i * 256;
    x[(size_t)r * HDIM + d] = (y[i] - mean) * rstd * g[d] + b[d];
  }
}

// x[m] = byte_emb[dec_in[m]] + pos_emb[m]; dec_in = [256, tb[0..30]]
__global__ __launch_bounds__(256) void embed_kernel(const int* __restrict__ tb,
                                                    const float* __restrict__ byte_emb,
                                                    const float* __restrict__ pos_emb,
                                                    float* __restrict__ x) {
  const int m = blockIdx.x, tid = threadIdx.x;
  const int idx = (m == 0) ? 256 : tb[m - 1];
  for (int d = tid; d < HDIM; d += 256)
    x[(size_t)m * HDIM + d] =
        byte_emb[(size_t)idx * HDIM + d] + pos_emb[(size_t)m * HDIM + d];
}

__global__ void softmax33(const float* __restrict__ logits, float* __restrict__ out) {
  __shared__ float sh[33];
  __shared__ float smax, ssum;
  const int tid = threadIdx.x;
  if (tid < 33) sh[tid] = logits[tid];
  __syncthreads();
  if (tid == 0) {
    float m = sh[0];
    for (int i = 1; i < 33; ++i) m = fmaxf(m, sh[i]);
    float s = 0.0f;
    for (int i = 0; i < 33; ++i) s += expf(sh[i] - m);
    smax = m;
    ssum = s;
  }
  __syncthreads();
  if (tid < 33) out[tid] = expf(sh[tid] - smax) / ssum;
}

// ---------------------------------------------------------------------------
extern "C" void kernel_launch(void* const* d_in, const int* in_sizes, int n_in,
                              void* d_out, int out_size, void* d_ws, size_t ws_size,
                              hipStream_t stream) {
  (void)in_sizes; (void)n_in; (void)out_size; (void)ws_size;
  const float* patch    = (const float*)d_in[0];
  const int*   tb       = (const int*)d_in[1];
  const float* sp_w1    = (const float*)d_in[2];
  const float* sp_b1    = (const float*)d_in[3];
  const float* sp_w2    = (const float*)d_in[4];
  const float* sp_b2    = (const float*)d_in[5];
  const float* byte_emb = (const float*)d_in[6];
  const float* pos_emb  = (const float*)d_in[7];
  const float* proj_w   = (const float*)d_in[8];
  const float* proj_b   = (const float*)d_in[9];
  const float* sa_in_w  = (const float*)d_in[10];
  const float* sa_in_b  = (const float*)d_in[11];
  const float* sa_out_w = (const float*)d_in[12];
  const float* sa_out_b = (const float*)d_in[13];
  const float* ca_in_w  = (const float*)d_in[14];
  const float* ca_in_b  = (const float*)d_in[15];
  const float* ca_out_w = (const float*)d_in[16];
  const float* ca_out_b = (const float*)d_in[17];
  const float* ff_w1    = (const float*)d_in[18];
  const float* ff_b1    = (const float*)d_in[19];
  const float* ff_w2    = (const float*)d_in[20];
  const float* ff_b2    = (const float*)d_in[21];
  const float* ln1_g    = (const float*)d_in[22];
  const float* ln1_b    = (const float*)d_in[23];
  const float* ln2_g    = (const float*)d_in[24];
  const float* ln2_b    = (const float*)d_in[25];
  const float* ln3_g    = (const float*)d_in[26];
  const float* ln3_b    = (const float*)d_in[27];

  float* out = (float*)d_out;
  float* ws  = (float*)d_ws;
  float* x    = ws;                  // 32*2048
  float* qkv  = x + 32 * 2048;       // 32*6144
  float* atto = qkv + 32 * 6144;     // 32*2048
  float* tmp  = atto + 32 * 2048;    // 32*2048
  float* ffh  = tmp + 32 * 2048;     // 32*8192
  float* hsp  = ffh + 32 * 8192;     // 1024
  float* lg33 = hsp + 1024;          // 33 (padded)
  float* vbuf = lg33 + 64;           // 2048
  float* cvec = vbuf + 2048;         // 2048

  const dim3 blk(256);

  // ---- size predictor: softmax(sp_w2 @ gelu(sp_w1 @ p + b1) + b2) ----
  gemv_wt<<<1024 / 8, blk, 0, stream>>>(patch, sp_w1, sp_b1, hsp, 2048, 1024, 1);
  gemv_wt<<<(33 + 7) / 8, blk, 0, stream>>>(hsp, sp_w2, sp_b2, lg33, 1024, 33, 0);
  softmax33<<<1, 64, 0, stream>>>(lg33, out);

  // ---- decoder input embedding ----
  embed_kernel<<<32, blk, 0, stream>>>(tb, byte_emb, pos_emb, x);

  for (int i = 0; i < NLAYER; ++i) {
    // self-attention
    gemm32_wmma<0><<<6144 / 128, blk, 0, stream>>>(
        x, sa_in_w + (size_t)i * 6144 * 2048, sa_in_b + i * 6144, qkv, 2048, 6144);
    attn_self<<<8, blk, 0, stream>>>(qkv, atto);
    gemm32_wmma<0><<<2048 / 128, blk, 0, stream>>>(
        atto, sa_out_w + (size_t)i * 2048 * 2048, sa_out_b + i * 2048, tmp, 2048, 2048);
    resid_ln<<<32, blk, 0, stream>>>(x, tmp, 2048, ln1_g + i * 2048, ln1_b + i * 2048);

    // cross-attention: single KV token => softmax == 1 => o == v (broadcast).
    // Only the wv third of ca_in_w and the output projection are needed.
    gemv_wt<<<2048 / 8, blk, 0, stream>>>(
        patch, ca_in_w + ((size_t)i * 3 * 2048 + 2 * 2048) * 2048,
        ca_in_b + i * 3 * 2048 + 2 * 2048, vbuf, 2048, 2048, 0);
    gemv_wt<<<2048 / 8, blk, 0, stream>>>(
        vbuf, ca_out_w + (size_t)i * 2048 * 2048, ca_out_b + i * 2048, cvec, 2048, 2048, 0);
    resid_ln<<<32, blk, 0, stream>>>(x, cvec, 0, ln2_g + i * 2048, ln2_b + i * 2048);

    // feed-forward
    gemm32_wmma<1><<<8192 / 128, blk, 0, stream>>>(
        x, ff_w1 + (size_t)i * 8192 * 2048, ff_b1 + i * 8192, ffh, 2048, 8192);
    gemm32_wmma<0><<<2048 / 128, blk, 0, stream>>>(
        ffh, ff_w2 + (size_t)i * 2048 * 8192, ff_b2 + i * 2048, tmp, 8192, 2048);
    resid_ln<<<32, blk, 0, stream>>>(x, tmp, 2048, ln3_g + i * 2048, ln3_b + i * 2048);
  }

  // ---- final byte logits -> out[33:] ----
  gemm32_wmma<0><<<256 / 128, blk, 0, stream>>>(x, proj_w, proj_b, out + 33, 2048, 256);
}